// GCN_30863634989386
// MI455X (gfx1250) — compile-verified
//
#include <hip/hip_runtime.h>
#include <math.h>

// ---------------------------------------------------------------------------
// GCN 2-layer forward for gfx1250 (MI455X), fp32 end-to-end.
// Dense GEMMs use V_WMMA_F32_16X16X4_F32 (exact fp32 matrix pipe).
// Edge aggregation is L2-resident scatter-add (192MB L2 holds all
// intermediates), so the kernel is L2-bandwidth/atomic bound; WMMA keeps the
// (cheap) GEMMs off the VALU.
// ---------------------------------------------------------------------------

#define N_NODES 100000
#define N_EDGES 1600000
#define DIN 128
#define DH 64
#define DOUT 16

typedef __attribute__((ext_vector_type(2))) float v2f;
typedef __attribute__((ext_vector_type(8))) float v8f;

// ---------------- degree / dinv ----------------

__global__ __launch_bounds__(256) void gcn_init_deg(float* __restrict__ deg) {
  int i = blockIdx.x * 256 + threadIdx.x;
  if (i < N_NODES) deg[i] = 1.0f;  // +1 self loop
}

__global__ __launch_bounds__(256) void gcn_deg_atomic(const int* __restrict__ dst,
                                                      float* __restrict__ deg) {
  int i = blockIdx.x * 256 + threadIdx.x;
  if (i < N_EDGES) atomicAdd(&deg[dst[i]], 1.0f);
}

__global__ __launch_bounds__(256) void gcn_rsqrt(float* __restrict__ deg) {
  int i = blockIdx.x * 256 + threadIdx.x;
  if (i < N_NODES) deg[i] = rsqrtf(deg[i]);
}

// ---------------- GEMM1: h1 = x @ W1 ; agg1 = h1*dinv^2 + b1 ----------------
// block = 128 threads = 4 waves. Block covers rows [16*bid, 16*bid+16),
// wave w covers cols [16w, 16w+16). K loop: 128/4 = 32 WMMA f32 16x16x4.
// A frag (16x4 f32): lane m = lane&15, VGPR j holds K = 2*(lane>>4)+j.
// B frag (4x16 f32): lane n = lane&15, VGPR j holds K = 2*(lane>>4)+j.
// C/D (16x16 f32):   col = lane&15, VGPR v holds row = v + 8*(lane>>4).

__global__ __launch_bounds__(128) void gcn_gemm1(
    const float* __restrict__ x, const float* __restrict__ W,
    const float* __restrict__ bias, const float* __restrict__ dinv,
    float* __restrict__ h, float* __restrict__ agg) {
  const int lane = threadIdx.x & 31;
  const int wave = threadIdx.x >> 5;
  const int row0 = blockIdx.x * 16;
  const int colB = wave * 16;
  const int mn = lane & 15;
  const int kh = lane >> 4;  // 0 or 1

  const float* __restrict__ xrow = x + (size_t)(row0 + mn) * DIN;

  v8f acc = {};
#pragma unroll
  for (int k = 0; k < DIN; k += 4) {
    const int kk = k + 2 * kh;
    v2f a, bf;
    a[0] = xrow[kk];
    a[1] = xrow[kk + 1];
    bf[0] = W[(size_t)kk * DH + colB + mn];
    bf[1] = W[(size_t)(kk + 1) * DH + colB + mn];
    acc = __builtin_amdgcn_wmma_f32_16x16x4_f32(
        /*neg_a=*/false, a, /*neg_b=*/false, bf,
        /*c_mod=*/(short)0, acc, /*reuse_a=*/false, /*reuse_b=*/false);
  }

  const int col = colB + mn;
  const float bv = bias[col];
#pragma unroll
  for (int v = 0; v < 8; ++v) {
    const int row = row0 + v + 8 * kh;
    const float val = acc[v];
    const float di = dinv[row];
    h[(size_t)row * DH + col] = val;
    agg[(size_t)row * DH + col] = val * di * di + bv;  // self-loop + bias
  }
}

// ---------------- edge scatter layer 1 (64 features, 1 wave / edge) --------

__global__ __launch_bounds__(256) void gcn_scatter1(
    const int* __restrict__ src, const int* __restrict__ dst,
    const float* __restrict__ dinv, const float* __restrict__ h,
    float* __restrict__ agg) {
  const int e = blockIdx.x * 8 + (threadIdx.x >> 5);
  const int lane = threadIdx.x & 31;
  if (e >= N_EDGES) return;
  const int s = src[e];
  const int d = dst[e];
  const float norm = dinv[s] * dinv[d];
  const float v0 = h[(size_t)s * DH + lane] * norm;
  const float v1 = h[(size_t)s * DH + lane + 32] * norm;
  atomicAdd(&agg[(size_t)d * DH + lane], v0);
  atomicAdd(&agg[(size_t)d * DH + lane + 32], v1);
}

// ---------------- ReLU in place ----------------

__global__ __launch_bounds__(256) void gcn_relu(float* __restrict__ a, int n) {
  int i = blockIdx.x * 256 + threadIdx.x;
  if (i < n) a[i] = fmaxf(a[i], 0.0f);
}

// ---------------- GEMM2: h2 = relu(agg1) @ W2 ; agg2 = h2*dinv^2 + b2 ------
// 1 wave = one 16x16 tile (DOUT == 16). 4 row-tiles per block.

__global__ __launch_bounds__(128) void gcn_gemm2(
    const float* __restrict__ in, const float* __restrict__ W,
    const float* __restrict__ bias, const float* __restrict__ dinv,
    float* __restrict__ h, float* __restrict__ agg) {
  const int lane = threadIdx.x & 31;
  const int wave = threadIdx.x >> 5;
  const int tile = blockIdx.x * 4 + wave;
  if (tile >= N_NODES / 16) return;  // wave-uniform: EXEC stays all-ones
  const int row0 = tile * 16;
  const int mn = lane & 15;
  const int kh = lane >> 4;

  const float* __restrict__ xrow = in + (size_t)(row0 + mn) * DH;

  v8f acc = {};
#pragma unroll
  for (int k = 0; k < DH; k += 4) {
    const int kk = k + 2 * kh;
    v2f a, bf;
    a[0] = xrow[kk];
    a[1] = xrow[kk + 1];
    bf[0] = W[(size_t)kk * DOUT + mn];
    bf[1] = W[(size_t)(kk + 1) * DOUT + mn];
    acc = __builtin_amdgcn_wmma_f32_16x16x4_f32(
        false, a, false, bf, (short)0, acc, false, false);
  }

  const float bv = bias[mn];
#pragma unroll
  for (int v = 0; v < 8; ++v) {
    const int row = row0 + v + 8 * kh;
    const float val = acc[v];
    const float di = dinv[row];
    h[(size_t)row * DOUT + mn] = val;
    agg[(size_t)row * DOUT + mn] = val * di * di + bv;
  }
}

// ---------------- edge scatter layer 2 (16 features, 16 lanes / edge) ------

__global__ __launch_bounds__(256) void gcn_scatter2(
    const int* __restrict__ src, const int* __restrict__ dst,
    const float* __restrict__ dinv, const float* __restrict__ h,
    float* __restrict__ agg) {
  const int gid = blockIdx.x * 256 + threadIdx.x;
  const int e = gid >> 4;
  const int f = gid & 15;
  if (e >= N_EDGES) return;
  const int s = src[e];
  const int d = dst[e];
  const float norm = dinv[s] * dinv[d];
  atomicAdd(&agg[(size_t)d * DOUT + f], h[(size_t)s * DOUT + f] * norm);
}

// ---------------- row-wise log_softmax over 16 cols ----------------

__global__ __launch_bounds__(256) void gcn_logsoftmax(
    const float* __restrict__ agg, float* __restrict__ out) {
  const int row = blockIdx.x * 256 + threadIdx.x;
  if (row >= N_NODES) return;
  const float* __restrict__ p = agg + (size_t)row * DOUT;
  float v[DOUT];
  float m = -INFINITY;
#pragma unroll
  for (int i = 0; i < DOUT; ++i) {
    v[i] = p[i];
    m = fmaxf(m, v[i]);
  }
  float s = 0.0f;
#pragma unroll
  for (int i = 0; i < DOUT; ++i) s += expf(v[i] - m);
  const float lse = m + logf(s);
#pragma unroll
  for (int i = 0; i < DOUT; ++i) out[(size_t)row * DOUT + i] = v[i] - lse;
}

// ---------------------------------------------------------------------------

extern "C" void kernel_launch(void* const* d_in, const int* in_sizes, int n_in,
                              void* d_out, int out_size, void* d_ws,
                              size_t ws_size, hipStream_t stream) {
  (void)in_sizes; (void)n_in; (void)out_size; (void)ws_size;

  const float* x  = (const float*)d_in[0];
  const int*   ei = (const int*)d_in[1];
  const float* W1 = (const float*)d_in[2];
  const float* b1 = (const float*)d_in[3];
  const float* W2 = (const float*)d_in[4];
  const float* b2 = (const float*)d_in[5];
  float* out = (float*)d_out;

  const int* src = ei;            // edge_index[0, :]
  const int* dst = ei + N_EDGES;  // edge_index[1, :]

  // Workspace carve-out (all fully initialized every call):
  float* ws   = (float*)d_ws;
  float* dinv = ws;                                   // N        (deg -> dinv)
  float* h1   = dinv + N_NODES;                       // N*64
  float* agg1 = h1 + (size_t)N_NODES * DH;            // N*64
  float* h2   = agg1 + (size_t)N_NODES * DH;          // N*16
  float* agg2 = h2 + (size_t)N_NODES * DOUT;          // N*16

  // 1) degrees (dst-counted, +1 self loop) -> dinv
  gcn_init_deg<<<(N_NODES + 255) / 256, 256, 0, stream>>>(dinv);
  gcn_deg_atomic<<<(N_EDGES + 255) / 256, 256, 0, stream>>>(dst, dinv);
  gcn_rsqrt<<<(N_NODES + 255) / 256, 256, 0, stream>>>(dinv);

  // 2) layer 1 GEMM (WMMA f32) + fused self-loop/bias init of agg1
  gcn_gemm1<<<N_NODES / 16, 128, 0, stream>>>(x, W1, b1, dinv, h1, agg1);

  // 3) edge aggregation layer 1
  gcn_scatter1<<<N_EDGES / 8, 256, 0, stream>>>(src, dst, dinv, h1, agg1);

  // 4) ReLU
  gcn_relu<<<(N_NODES * DH + 255) / 256, 256, 0, stream>>>(agg1, N_NODES * DH);

  // 5) layer 2 GEMM (WMMA f32) + fused self-loop/bias init of agg2
  gcn_gemm2<<<(N_NODES / 16 + 3) / 4, 128, 0, stream>>>(agg1, W2, b2, dinv, h2,
                                                        agg2);

  // 6) edge aggregation layer 2
  gcn_scatter2<<<(N_EDGES * 16 + 255) / 256, 256, 0, stream>>>(src, dst, dinv,
                                                               h2, agg2);

  // 7) log_softmax -> output
  gcn_logsoftmax<<<(N_NODES + 255) / 256, 256, 0, stream>>>(agg2, out);
}